// ConformerModel_69810398429358
// MI455X (gfx1250) — compile-verified
//
#include <hip/hip_runtime.h>

typedef float v2f __attribute__((ext_vector_type(2)));
typedef float v4f __attribute__((ext_vector_type(4)));
typedef float v8f __attribute__((ext_vector_type(8)));

#define B_ 4
#define T_ 256
#define U_ 64
#define D_ 512
#define C_ 1024
#define LDW_ (2 * D_)

#define WMMA_F32(a, b, c) \
    __builtin_amdgcn_wmma_f32_16x16x4_f32(false, (a), false, (b), (short)0, (c), false, false)

// ---------------------------------------------------------------------------
// GEMM: out[m, c] = sum_d A[m, d] * W[c, d]   (A: MxD_ row-major, W stride LDW_)
// One wave computes a 16x64 tile via V_WMMA_F32_16X16X4_F32, K-loop software-
// pipelined one step ahead so VMEM latency overlaps the WMMA chain.
// A-frag (16x4 f32): lane L holds (A[M=L&15][koff], A[M=L&15][koff+1]), koff=2*(L>>4)
// B-frag (4x16 f32): lane L holds (B[koff][N=L&15], B[koff+1][N=L&15])
// D (16x16 f32):     elem v of v8f is D[M = 8*(L>>4) + v][N = L&15]
// ---------------------------------------------------------------------------
__global__ __launch_bounds__(256) void gemm_wmma_f32(
    const float* __restrict__ A, const float* __restrict__ W,
    float* __restrict__ out, int M)
{
    const int lane = threadIdx.x & 31;
    const int wave = blockIdx.x * (blockDim.x >> 5) + (threadIdx.x >> 5);
    const int cstrips = C_ >> 6;                // 64-wide column strips
    const int ntiles  = (M >> 4) * cstrips;
    if (wave >= ntiles) return;                 // whole-wave uniform: EXEC stays all-1s

    const int m0 = (wave / cstrips) << 4;
    const int c0 = (wave % cstrips) << 6;

    const int r    = lane & 15;
    const int half = lane >> 4;
    const int koff = half << 1;

    const float* aRow  = A + (size_t)(m0 + r) * D_ + koff;
    const float* wRow0 = W + (size_t)(c0 +  0 + r) * LDW_ + koff;
    const float* wRow1 = W + (size_t)(c0 + 16 + r) * LDW_ + koff;
    const float* wRow2 = W + (size_t)(c0 + 32 + r) * LDW_ + koff;
    const float* wRow3 = W + (size_t)(c0 + 48 + r) * LDW_ + koff;

    v8f acc0 = {}, acc1 = {}, acc2 = {}, acc3 = {};

    // ---- software pipeline: preload step 0 ----
    v2f a_c  = *(const v2f*)(aRow);
    v2f b0_c = *(const v2f*)(wRow0);
    v2f b1_c = *(const v2f*)(wRow1);
    v2f b2_c = *(const v2f*)(wRow2);
    v2f b3_c = *(const v2f*)(wRow3);

    #pragma unroll 4
    for (int d = 0; d < D_ - 4; d += 4) {
        // prefetch next K-step before consuming the current one
        const int dn = d + 4;
        v2f a_n  = *(const v2f*)(aRow  + dn);
        v2f b0_n = *(const v2f*)(wRow0 + dn);
        v2f b1_n = *(const v2f*)(wRow1 + dn);
        v2f b2_n = *(const v2f*)(wRow2 + dn);
        v2f b3_n = *(const v2f*)(wRow3 + dn);

        acc0 = WMMA_F32(a_c, b0_c, acc0);
        acc1 = WMMA_F32(a_c, b1_c, acc1);
        acc2 = WMMA_F32(a_c, b2_c, acc2);
        acc3 = WMMA_F32(a_c, b3_c, acc3);

        a_c = a_n; b0_c = b0_n; b1_c = b1_n; b2_c = b2_n; b3_c = b3_n;
    }
    // ---- epilogue step ----
    acc0 = WMMA_F32(a_c, b0_c, acc0);
    acc1 = WMMA_F32(a_c, b1_c, acc1);
    acc2 = WMMA_F32(a_c, b2_c, acc2);
    acc3 = WMMA_F32(a_c, b3_c, acc3);

    float* o = out + (size_t)(m0 + (half << 3)) * C_ + c0 + r;
    #pragma unroll
    for (int v = 0; v < 8; ++v) {
        float* orow = o + (size_t)v * C_;
        orow[ 0] = acc0[v];
        orow[16] = acc1[v];
        orow[32] = acc2[v];
        orow[48] = acc3[v];
    }
}

// ---------------------------------------------------------------------------
// Broadcast add: out[bt, u, c] = enc[bt, c] + dec[b*U + u, c]
// This is the roofline-dominant stream: 256 MB of output at 23.3 TB/s ~ 11.5 us.
// float4-vectorized over c; each thread handles 8 consecutive u values so the
// enc value is read once from L2 per 8 output rows. Non-temporal stores keep
// the output stream from evicting the 5 MB of projections in L2.
// ---------------------------------------------------------------------------
#define UPT 8
__global__ __launch_bounds__(256) void bcast_add(
    const float* __restrict__ enc, const float* __restrict__ dec,
    float* __restrict__ out)
{
    const int C4 = C_ / 4;
    const long long i = (long long)blockIdx.x * blockDim.x + threadIdx.x;

    const int c4  = (int)(i % C4);
    long long t1  = i / C4;
    const int uc  = (int)(t1 % (U_ / UPT));
    const int bt  = (int)(t1 / (U_ / UPT));
    const int b   = bt / T_;
    const int u0  = uc * UPT;

    const v4f e = *(const v4f*)(enc + (size_t)bt * C_ + c4 * 4);

    const float* dbase = dec + ((size_t)b * U_ + u0) * C_ + c4 * 4;
    float* obase = out + (((size_t)bt * U_ + u0) * C_) + c4 * 4;

    #pragma unroll
    for (int k = 0; k < UPT; ++k) {
        v4f d = *(const v4f*)(dbase + (size_t)k * C_);
        v4f s = e + d;
        __builtin_nontemporal_store(s, (v4f*)(obase + (size_t)k * C_));
    }
}

extern "C" void kernel_launch(void* const* d_in, const int* in_sizes, int n_in,
                              void* d_out, int out_size, void* d_ws, size_t ws_size,
                              hipStream_t stream) {
    const float* enc_in = (const float*)d_in[0];   // (B, T, D)
    const float* dec_in = (const float*)d_in[1];   // (B, U, D)
    const float* W      = (const float*)d_in[2];   // (C, 2D)
    float* out = (float*)d_out;                    // (B, T, U, C)

    float* enc_proj = (float*)d_ws;                               // (B*T, C): 4 MB
    float* dec_proj = enc_proj + (size_t)B_ * T_ * C_;            // (B*U, C): 1 MB

    // enc projection: M = B*T = 1024 -> 64*16 = 1024 wave-tiles -> 128 blocks of 8 waves
    {
        int M = B_ * T_;
        int tiles = (M >> 4) * (C_ >> 6);
        gemm_wmma_f32<<<dim3((tiles + 7) / 8), 256, 0, stream>>>(enc_in, W, enc_proj, M);
    }
    // dec projection: M = B*U = 256 -> 256 wave-tiles -> 32 blocks
    {
        int M = B_ * U_;
        int tiles = (M >> 4) * (C_ >> 6);
        gemm_wmma_f32<<<dim3((tiles + 7) / 8), 256, 0, stream>>>(dec_in, W + D_, dec_proj, M);
    }
    // broadcast add: B*T*(U/8)*C4 threads = 2,097,152 -> 8192 blocks of 256
    {
        long long nthreads = (long long)B_ * T_ * (U_ / UPT) * (C_ / 4);
        bcast_add<<<dim3((unsigned)(nthreads / 256)), 256, 0, stream>>>(
            enc_proj, dec_proj, out);
    }
}